// DKVMN_21320217657933
// MI455X (gfx1250) — compile-verified
//
#include <hip/hip_runtime.h>
#include <math.h>

typedef __attribute__((ext_vector_type(2))) float v2f;
typedef __attribute__((ext_vector_type(8))) float v8f;

#define NUM_Q_ 10000
#define B_    64
#define S_    200
#define DK_   128
#define DV_   64
#define BS_   (B_ * S_)

__device__ __forceinline__ v8f wmma_f32(v2f a, v2f b, v8f c) {
    // D = A(16x4) * B(4x16) + C(16x16), fp32, wave32
    return __builtin_amdgcn_wmma_f32_16x16x4_f32(false, a, false, b,
                                                 (short)0, c, false, false);
}

__device__ __forceinline__ float sigmoidf_(float x) { return 1.0f / (1.0f + expf(-x)); }

// ---------------------------------------------------------------------------
// Kernel 0: transpose weight matrices once so WMMA B-operands are contiguous
// along K.  eWt/aWt: [N=128][K=128], fWt: [N=128][K=256].
// ---------------------------------------------------------------------------
__global__ __launch_bounds__(256) void dkvmn_transpose(
    const float* __restrict__ eW, const float* __restrict__ aW,
    const float* __restrict__ fW,
    float* __restrict__ eWt, float* __restrict__ aWt, float* __restrict__ fWt)
{
    int idx = blockIdx.x * 256 + threadIdx.x;      // 0 .. 32767
    if (idx < DK_ * DK_) {
        int n = idx >> 7, kx = idx & (DK_ - 1);
        eWt[idx] = eW[kx * DK_ + n];
        aWt[idx] = aW[kx * DK_ + n];
    }
    {
        int n = idx >> 8, kx = idx & 255;          // fWt[n][k], k < 256
        fWt[idx] = fW[kx * DK_ + n];
    }
}

// ---------------------------------------------------------------------------
// Kernel 1: gather k/v, compute w = softmax(k Mk^T), e = sigmoid(v eW + eb),
//           a = tanh(v aW + ab).  One block = 16 (b,s) rows, 4 waves.
// ---------------------------------------------------------------------------
__global__ __launch_bounds__(128) void dkvmn_embed_gate(
    const int* __restrict__ qseq, const int* __restrict__ cseq,
    const float* __restrict__ key_emb, const float* __restrict__ value_emb,
    const float* __restrict__ Mk,
    const float* __restrict__ eWt, const float* __restrict__ ebias,
    const float* __restrict__ aWt, const float* __restrict__ abias,
    float* __restrict__ w_out, float* __restrict__ e_out, float* __restrict__ a_out)
{
    __shared__ float4 sKbuf[16 * 32];   // 16 rows x 128 f32
    __shared__ float4 sVbuf[16 * 32];
    __shared__ float  sL[16 * DV_];     // logits for softmax
    float* sK = (float*)sKbuf;
    float* sV = (float*)sVbuf;

    const int tid  = threadIdx.x;
    const int lane = tid & 31;
    const int wave = tid >> 5;
    const int row0 = blockIdx.x * 16;

    // stage gathered embedding rows (float4 vector loads)
    for (int slot = tid; slot < 16 * 32; slot += 128) {
        int r = slot >> 5, c4 = slot & 31;
        int bs = row0 + r;
        int q  = qseq[bs];
        int cc = cseq[bs];
        sKbuf[r * 32 + c4] = ((const float4*)(key_emb   + (size_t)q * DK_))[c4];
        sVbuf[r * 32 + c4] = ((const float4*)(value_emb + ((size_t)q + (size_t)NUM_Q_ * cc) * DK_))[c4];
    }
    __syncthreads();

    const int m  = lane & 15;   // A row
    const int hi = lane >> 4;   // lane half selects K pair
    const int n  = lane & 15;   // B/C column

    // 20 n-tile jobs: 0..3 logits (N=64), 4..11 e (N=128), 12..19 a (N=128)
    for (int job = wave; job < 20; job += 4) {
        const float* sA;
        const float* bmat;
        int nbase, kind;
        if (job < 4)       { kind = 0; sA = sK; nbase = job * 16;        bmat = Mk;  }
        else if (job < 12) { kind = 1; sA = sV; nbase = (job - 4) * 16;  bmat = eWt; }
        else               { kind = 2; sA = sV; nbase = (job - 12) * 16; bmat = aWt; }

        const float* arow = sA + m * DK_;
        const float* bptr = bmat + (size_t)(nbase + n) * DK_;   // contiguous along K
        v8f c = {};
        for (int kk = 0; kk < DK_; kk += 4) {
            int kb = kk + 2 * hi;
            v2f a = *(const v2f*)(arow + kb);
            v2f b = *(const v2f*)(bptr + kb);
            c = wmma_f32(a, b, c);
        }

        if (kind == 0) {
            #pragma unroll
            for (int r = 0; r < 8; ++r) {
                int mm = r + 8 * hi;
                sL[mm * DV_ + nbase + n] = c[r];
            }
        } else {
            const float* bias = (kind == 1) ? ebias : abias;
            float* outp       = (kind == 1) ? e_out : a_out;
            #pragma unroll
            for (int r = 0; r < 8; ++r) {
                int mm = r + 8 * hi;
                float x = c[r] + bias[nbase + n];
                float y = (kind == 1) ? sigmoidf_(x) : tanhf(x);
                outp[(size_t)(row0 + mm) * DK_ + nbase + n] = y;
            }
        }
    }
    __syncthreads();

    // softmax over DV per row
    if (tid < 16) {
        int bs = row0 + tid;
        float mx = -1e30f;
        for (int v = 0; v < DV_; ++v) mx = fmaxf(mx, sL[tid * DV_ + v]);
        float s = 0.0f;
        for (int v = 0; v < DV_; ++v) {
            float ex = expf(sL[tid * DV_ + v] - mx);
            sL[tid * DV_ + v] = ex; s += ex;
        }
        float inv = 1.0f / s;
        for (int v = 0; v < DV_; ++v)
            w_out[(size_t)bs * DV_ + v] = sL[tid * DV_ + v] * inv;
    }
}

// ---------------------------------------------------------------------------
// Kernel 2: sequential memory scan. One block per batch element; thread k
// holds Mv column k (64 f32) entirely in VGPRs.  Double-buffered w broadcast:
// one barrier per step; float4 LDS broadcast reads.
// ---------------------------------------------------------------------------
__global__ __launch_bounds__(128) void dkvmn_scan(
    const float* __restrict__ w, const float* __restrict__ e,
    const float* __restrict__ a, const float* __restrict__ Mv0,
    float* __restrict__ reads)
{
    __shared__ float wbuf[2][DV_];
    const int b = blockIdx.x;
    const int k = threadIdx.x;      // 0..127

    float Mv[DV_];
    #pragma unroll
    for (int v = 0; v < DV_; ++v) Mv[v] = Mv0[v * DK_ + k];

    const float* wb = w + (size_t)b * S_ * DV_;
    const float* ep = e + (size_t)b * S_ * DK_;
    const float* ap = a + (size_t)b * S_ * DK_;
    float*       rp = reads + (size_t)b * S_ * DK_;

    if (k < DV_) wbuf[0][k] = wb[k];
    __syncthreads();

    for (int t = 0; t < S_; ++t) {
        const int p = t & 1;
        if ((t + 1 < S_) && (k < DV_)) wbuf[p ^ 1][k] = wb[(t + 1) * DV_ + k];
        float ek = ep[t * DK_ + k];
        float ak = ap[t * DK_ + k];
        float acc = 0.0f;
        const float4* w4 = (const float4*)wbuf[p];
        #pragma unroll
        for (int v4 = 0; v4 < DV_ / 4; ++v4) {
            float4 wv = w4[v4];
            #pragma unroll
            for (int j = 0; j < 4; ++j) {
                int v = v4 * 4 + j;
                float wj = (j == 0) ? wv.x : (j == 1) ? wv.y : (j == 2) ? wv.z : wv.w;
                acc   = fmaf(wj, Mv[v], acc);                    // read (pre-update)
                Mv[v] = fmaf(wj, fmaf(-ek, Mv[v], ak), Mv[v]);   // Mv += w*(a - e*Mv)
            }
        }
        rp[t * DK_ + k] = acc;
        __syncthreads();
    }
}

// ---------------------------------------------------------------------------
// Kernel 3: f = tanh([reads, k] @ f_W + f_b);  out = sigmoid(f . p_W + p_b)
// ---------------------------------------------------------------------------
__global__ __launch_bounds__(128) void dkvmn_readout(
    const int* __restrict__ qseq, const float* __restrict__ key_emb,
    const float* __restrict__ reads,
    const float* __restrict__ fWt, const float* __restrict__ fbias,
    const float* __restrict__ pW, const float* __restrict__ pbias,
    float* __restrict__ out)
{
    __shared__ float4 sXbuf[16 * 64];   // 16 rows x 256 f32 (reads || k)
    __shared__ float  sF[16 * DK_];
    float* sX = (float*)sXbuf;

    const int tid  = threadIdx.x;
    const int lane = tid & 31;
    const int wave = tid >> 5;
    const int row0 = blockIdx.x * 16;

    for (int slot = tid; slot < 16 * 32; slot += 128) {
        int r = slot >> 5, c4 = slot & 31;
        int bs = row0 + r;
        sXbuf[r * 64 + c4]      = ((const float4*)(reads + (size_t)bs * DK_))[c4];
        int q = qseq[bs];
        sXbuf[r * 64 + 32 + c4] = ((const float4*)(key_emb + (size_t)q * DK_))[c4];
    }
    __syncthreads();

    const int m  = lane & 15;
    const int hi = lane >> 4;
    const int n  = lane & 15;

    for (int job = wave; job < 8; job += 4) {
        int nbase = job * 16;
        const float* arow = sX + m * 256;
        const float* bptr = fWt + (size_t)(nbase + n) * 256;  // contiguous along K
        v8f c = {};
        for (int kk = 0; kk < 2 * DK_; kk += 4) {
            int kb = kk + 2 * hi;
            v2f a = *(const v2f*)(arow + kb);
            v2f b = *(const v2f*)(bptr + kb);
            c = wmma_f32(a, b, c);
        }
        #pragma unroll
        for (int r = 0; r < 8; ++r) {
            int mm = r + 8 * hi;
            sF[mm * DK_ + nbase + n] = tanhf(c[r] + fbias[nbase + n]);
        }
    }
    __syncthreads();

    if (tid < 16) {
        float acc = pbias[0];
        for (int nn = 0; nn < DK_; ++nn)
            acc = fmaf(sF[tid * DK_ + nn], pW[nn], acc);
        out[row0 + tid] = sigmoidf_(acc);
    }
}

// ---------------------------------------------------------------------------
extern "C" void kernel_launch(void* const* d_in, const int* in_sizes, int n_in,
                              void* d_out, int out_size, void* d_ws, size_t ws_size,
                              hipStream_t stream) {
    const int*   qseq      = (const int*)d_in[0];
    const int*   cseq      = (const int*)d_in[1];
    const float* key_emb   = (const float*)d_in[2];
    const float* value_emb = (const float*)d_in[3];
    const float* Mk        = (const float*)d_in[4];
    const float* Mv0       = (const float*)d_in[5];
    const float* fW        = (const float*)d_in[6];
    const float* fb        = (const float*)d_in[7];
    const float* eW        = (const float*)d_in[8];
    const float* ebv       = (const float*)d_in[9];
    const float* aW        = (const float*)d_in[10];
    const float* abv       = (const float*)d_in[11];
    const float* pW        = (const float*)d_in[12];
    const float* pb        = (const float*)d_in[13];
    float* out = (float*)d_out;

    float* ws    = (float*)d_ws;
    float* ws_w  = ws;                              // BS * DV
    float* ws_e  = ws_w  + (size_t)BS_ * DV_;       // BS * DK
    float* ws_a  = ws_e  + (size_t)BS_ * DK_;       // BS * DK
    float* ws_r  = ws_a  + (size_t)BS_ * DK_;       // BS * DK
    float* ws_eWt = ws_r  + (size_t)BS_ * DK_;      // 128*128
    float* ws_aWt = ws_eWt + (size_t)DK_ * DK_;     // 128*128
    float* ws_fWt = ws_aWt + (size_t)DK_ * DK_;     // 128*256

    dkvmn_transpose<<<128, 256, 0, stream>>>(eW, aW, fW, ws_eWt, ws_aWt, ws_fWt);

    dkvmn_embed_gate<<<BS_ / 16, 128, 0, stream>>>(
        qseq, cseq, key_emb, value_emb, Mk, ws_eWt, ebv, ws_aWt, abv, ws_w, ws_e, ws_a);

    dkvmn_scan<<<B_, 128, 0, stream>>>(ws_w, ws_e, ws_a, Mv0, ws_r);

    dkvmn_readout<<<BS_ / 16, 128, 0, stream>>>(
        qseq, key_emb, ws_r, ws_fWt, fb, pW, pb, out);
}